// Conv2dManual_77695958385099
// MI455X (gfx1250) — compile-verified
//
#include <hip/hip_runtime.h>

// CDNA5 (gfx1250) implicit-GEMM 3x3 conv using V_WMMA_F32_16X16X4_F32.
// M = C_OUT(128), N = pixels, K = 9*64 = 576 ((kh,kw)-major, ci-minor).
// All 128x576 fp32 weights staged in 290KB LDS (320KB/WGP on CDNA5).
// v2: branch-free OOB handling (clamped loads + cndmask selects),
//     4 output rows per workgroup to amortize the weight staging.

typedef __attribute__((ext_vector_type(2))) float v2f;
typedef __attribute__((ext_vector_type(8))) float v8f;

#define KW 3
#define CIN 64
#define COUT 128
#define HH 256
#define WW 256
#define NBATCH 16
#define ROWS_PER_WG 4
#define KTOT (CIN * KW * KW)      /* 576 */
#define LDS_STRIDE 580            /* 576 + 4 dword pad: conflict-free ds_load_b64 */
#define LDS_FLOATS (COUT * LDS_STRIDE)

__global__ __launch_bounds__(256)
void conv2d_wmma_f32(const float* __restrict__ x,
                     const float* __restrict__ wgt,
                     const float* __restrict__ bias,
                     float* __restrict__ out)
{
    extern __shared__ float lw[];
    const int tid = threadIdx.x;
    const int h0  = blockIdx.x * ROWS_PER_WG;  // first output row of this WG
    const int n   = blockIdx.y;                // batch image

    // Stage weights OIHW -> LDS layout [co][(kh*3+kw)*64 + ci], stride LDS_STRIDE.
    // Global reads are fully linear/coalesced; done once per workgroup,
    // amortized over ROWS_PER_WG output rows.
    for (int idx = tid; idx < COUT * KTOT; idx += 256) {
        int co  = idx / KTOT;
        int rem = idx - co * KTOT;
        int ci  = rem / 9;
        int kk  = rem - ci * 9;
        lw[co * LDS_STRIDE + kk * CIN + ci] = wgt[idx];
    }
    __syncthreads();

    const int wave  = tid >> 5;
    const int lane  = tid & 31;
    const int ln15  = lane & 15;   // A: M index in block; B/C/D: N (pixel) index
    const int half  = lane >> 4;   // K-half selector (K={0,1} vs K={2,3})
    const int mbase = wave * 16;   // this wave's 16 output channels

    // Fold bias into accumulator init. C/D layout: VGPR r -> M = r + 8*half.
    float binit[8];
#pragma unroll
    for (int r = 0; r < 8; ++r)
        binit[r] = bias[mbase + r + 8 * half];

    // A-fragment base in LDS for this lane: row (mbase+ln15), K offset 2*half.
    const float* lwbase = lw + (size_t)(mbase + ln15) * LDS_STRIDE + 2 * half;

    const size_t HWs = (size_t)HH * WW;
    const float* ximg = x + (size_t)n * CIN * HWs;

    for (int hi = 0; hi < ROWS_PER_WG; ++hi) {
        const int h = h0 + hi;

        for (int p0 = 0; p0 < WW; p0 += 32) {
            v8f c0, c1;
#pragma unroll
            for (int r = 0; r < 8; ++r) { c0[r] = binit[r]; c1[r] = binit[r]; }

            for (int kh = 0; kh < KW; ++kh) {
                const int hr = h + kh - 1;
                if (hr < 0 || hr >= HH) continue;   // uniform: EXEC stays all-1s
                for (int kwp = 0; kwp < KW; ++kwp) {
                    // Column validity & clamped columns: loop-invariant over cs,
                    // loads are always in-bounds; OOB lanes zeroed by cndmask.
                    const int  w0  = p0 + ln15 + kwp - 1;  // tile-0 column
                    const int  w1  = w0 + 16;              // tile-1 column
                    const bool v0  = ((unsigned)w0) < WW;
                    const bool v1  = w1 < WW;              // w1 >= 15 always
                    const int  wc0 = v0 ? w0 : 0;
                    const int  wc1 = v1 ? w1 : 0;
                    const float* lwp = lwbase + (kh * 3 + kwp) * CIN;
                    const float* xp  = ximg + (size_t)(2 * half) * HWs
                                            + (size_t)hr * WW;
#pragma unroll 8
                    for (int cs = 0; cs < CIN / 4; ++cs) {
                        // A: {W[m][k], W[m][k+1]} -- ds_load_b64, conflict-free
                        v2f a = *(const v2f*)(lwp + cs * 4);
                        // B: unconditional coalesced loads from clamped columns
                        const float* xc = xp + (size_t)(cs * 4) * HWs;
                        float t00 = xc[wc0];
                        float t01 = xc[HWs + wc0];
                        float t10 = xc[wc1];
                        float t11 = xc[HWs + wc1];
                        v2f b0, b1;
                        b0.x = v0 ? t00 : 0.0f;
                        b0.y = v0 ? t01 : 0.0f;
                        b1.x = v1 ? t10 : 0.0f;
                        b1.y = v1 ? t11 : 0.0f;
                        c0 = __builtin_amdgcn_wmma_f32_16x16x4_f32(
                                 false, a, false, b0, (short)0, c0, false, false);
                        c1 = __builtin_amdgcn_wmma_f32_16x16x4_f32(
                                 false, a, false, b1, (short)0, c1, false, false);
                    }
                }
            }

            // Store: cout = mbase + r + 8*half, pixel = p0 + ln15 (+16 tile 1).
            float* op = out + ((size_t)n * COUT + mbase + 8 * half) * HWs
                            + (size_t)h * WW + p0 + ln15;
#pragma unroll
            for (int r = 0; r < 8; ++r) {
                op[(size_t)r * HWs]      = c0[r];
                op[(size_t)r * HWs + 16] = c1[r];
            }
        }
    }
}

extern "C" void kernel_launch(void* const* d_in, const int* in_sizes, int n_in,
                              void* d_out, int out_size, void* d_ws, size_t ws_size,
                              hipStream_t stream) {
    (void)in_sizes; (void)n_in; (void)d_ws; (void)ws_size; (void)out_size;
    const float* x    = (const float*)d_in[0];
    const float* wgt  = (const float*)d_in[1];
    const float* bias = (const float*)d_in[2];
    float* out        = (float*)d_out;

    dim3 grid(HH / ROWS_PER_WG, NBATCH);   // one WG per (4-row band, image)
    size_t ldsBytes = (size_t)LDS_FLOATS * sizeof(float);  // ~290 KB of 320 KB
    conv2d_wmma_f32<<<grid, 256, ldsBytes, stream>>>(x, wgt, bias, out);
}